// DecodePredictionsSoft_26525718020109
// MI455X (gfx1250) — compile-verified
//
#include <hip/hip_runtime.h>
#include <hip/hip_bf16.h>
#include <math.h>

#define NBATCH 4
#define NANCH  49104
#define NCLS   16
#define TSEL   100      // MAX_PER_CLASS
#define MAXDET 100
#define CONF_T 0.05f
#define IOU_T  0.5f
#define SIGMA  0.05f
#define NEGV   (-1.0e10f)
#define CAND_MAX 4096
#define DYN_LDS  ((size_t)CAND_MAX * (16 + 4 + 4))   // box4 + score + idx = 98304 B

// ---------------------------------------------------------------------------
// Kernel 1: decode boxes + per-anchor max class / max score (streaming, HBM-bound)
// ---------------------------------------------------------------------------
__global__ __launch_bounds__(256) void decode_kernel(
    const float* __restrict__ pred,     // [B, N, 20]
    const float* __restrict__ anchors,  // [N, 4]  (cx, cy, w, h)
    float* __restrict__ boxes,          // [B, N, 4] corners
    float* __restrict__ mscore,         // [B, N]
    int*   __restrict__ mcls)           // [B, N]
{
    int i = blockIdx.x * 256 + threadIdx.x;
    if (i >= NBATCH * NANCH) return;
    int n = i % NANCH;
    const float* p = pred + (size_t)i * 20;
    // CDNA5: global_prefetch_b8 of a future tile (speculative)
    __builtin_prefetch(p + 20 * 2048, 0, 0);

    float4 a = *(const float4*)(anchors + (size_t)n * 4);  // cx cy w h
    float cx = p[0] * a.z + a.x;
    float cy = p[1] * a.w + a.y;
    float w  = expf(p[2]) * a.z;
    float h  = expf(p[3]) * a.w;
    float4 bx = make_float4(cx - 0.5f * w, cy - 0.5f * h,
                            cx + 0.5f * w, cy + 0.5f * h);
    *(float4*)(boxes + (size_t)i * 4) = bx;

    float ml = p[4];
    int   mc = 0;
#pragma unroll
    for (int k = 1; k < NCLS; ++k) {
        float v = p[4 + k];
        if (v > ml) { ml = v; mc = k; }   // strict '>' == argmax first-max tiebreak
    }
    mscore[i] = 1.0f / (1.0f + expf(-ml));
    mcls[i]   = mc;
}

// ---------------------------------------------------------------------------
// Kernel 2: per-(batch,class) soft-NMS, candidates resident in LDS.
//   Phase 1: deterministic wave32 prefix compaction, 4 anchors/thread/tile
//            (float4/int4 loads), candidate boxes gathered memory->LDS with
//            the CDNA5 async path (global_load_async_to_lds_b128, ASYNCcnt).
//   Phase 2: 100 rounds of block-argmax (smallest-index tiebreak) + Gaussian
//            decay / hard suppression, all in LDS.
// ---------------------------------------------------------------------------
__global__ __launch_bounds__(256) void nms_kernel(
    const float* __restrict__ boxes,    // [B, N, 4]
    const float* __restrict__ mscore,   // [B, N]
    const int*   __restrict__ mcls,     // [B, N]
    int*   __restrict__ sel_idx,        // [B*C, T]
    float* __restrict__ sel_score,      // [B*C, T]
    int*   __restrict__ sel_pick)       // [B*C, T]
{
    extern __shared__ unsigned char dynsmem[];
    float4* s_box = (float4*)dynsmem;                                   // CAND_MAX
    float*  s_sc  = (float*) (dynsmem + (size_t)CAND_MAX * 16);         // CAND_MAX
    int*    s_ix  = (int*)   (dynsmem + (size_t)CAND_MAX * 20);         // CAND_MAX

    __shared__ int    s_wcnt[8];
    __shared__ int    s_wbase[8];
    __shared__ int    s_total;
    __shared__ float  s_rv[8];
    __shared__ int    s_rj[8];
    __shared__ float  s_selv;
    __shared__ int    s_selj;
    __shared__ float4 s_selbox;

    const int bc   = blockIdx.x;
    const int b    = bc / NCLS;
    const int c    = bc % NCLS;
    const int tid  = threadIdx.x;
    const int lane = tid & 31;
    const int wid  = tid >> 5;

    if (tid == 0) s_total = 0;
    __syncthreads();

    // low 32 bits of the flat shared-aperture address == LDS byte offset
    const unsigned sbox_lds = (unsigned)(size_t)(void*)s_box;
    const size_t bbase = (size_t)b * NANCH;

    // ---- Phase 1: ordered compaction + async gather of boxes into LDS ----
    // 4 consecutive anchors per thread per tile; NANCH % 4 == 0, so each quad
    // is fully in-range or fully out; float4/int4 loads stay 16B-aligned.
    for (int base = 0; base < NANCH; base += 1024) {
        int n0 = base + tid * 4;
        bool ok0 = false, ok1 = false, ok2 = false, ok3 = false;
        float4 scv = make_float4(0.f, 0.f, 0.f, 0.f);
        if (n0 < NANCH) {
            scv      = *(const float4*)(mscore + bbase + n0);
            int4 clv = *(const int4*)  (mcls   + bbase + n0);
            ok0 = (clv.x == c) && (scv.x >= CONF_T);
            ok1 = (clv.y == c) && (scv.y >= CONF_T);
            ok2 = (clv.z == c) && (scv.z >= CONF_T);
            ok3 = (clv.w == c) && (scv.w >= CONF_T);
            // prefetch next tile (global_prefetch_b8)
            __builtin_prefetch(mscore + bbase + n0 + 1024, 0, 0);
            __builtin_prefetch(mcls   + bbase + n0 + 1024, 0, 0);
        }
        int cnt = (int)ok0 + (int)ok1 + (int)ok2 + (int)ok3;
        // wave32 inclusive prefix sum of per-thread counts (tid order ==
        // anchor-index order, so the candidate list stays index-sorted)
        int incl = cnt;
#pragma unroll
        for (int d = 1; d < 32; d <<= 1) {
            int v = __shfl_up(incl, d, 32);
            if (lane >= d) incl += v;
        }
        if (lane == 31) s_wcnt[wid] = incl;
        __syncthreads();
        if (tid == 0) {
            int tot = s_total;
#pragma unroll
            for (int w = 0; w < 8; ++w) { s_wbase[w] = tot; tot += s_wcnt[w]; }
            s_total = tot;
        }
        __syncthreads();
        int pos = s_wbase[wid] + incl - cnt;    // exclusive prefix position
        const float sce[4] = { scv.x, scv.y, scv.z, scv.w };
        const bool  oke[4] = { ok0, ok1, ok2, ok3 };
#pragma unroll
        for (int e = 0; e < 4; ++e) {
            if (oke[e]) {
                if (pos < CAND_MAX) {
                    int n = n0 + e;
                    s_ix[pos] = n;
                    s_sc[pos] = sce[e];
                    unsigned laddr = sbox_lds + (unsigned)pos * 16u;
                    unsigned long long gaddr =
                        (unsigned long long)(size_t)(boxes + (bbase + (size_t)n) * 4);
                    // CDNA5 async-tensor path: memory -> LDS per-lane gather,
                    // no VGPR round trip, tracked by ASYNCcnt.
                    asm volatile("global_load_async_to_lds_b128 %0, %1, off"
                                 :: "v"(laddr), "v"(gaddr) : "memory");
                }
                ++pos;
            }
        }
    }
    asm volatile("s_wait_asynccnt 0x0" ::: "memory");
    __syncthreads();

    const int M = (s_total < CAND_MAX) ? s_total : CAND_MAX;
    const float scale = -0.5f / SIGMA;
    const int outb = bc * TSEL;

    // ---- Phase 2: iterative soft-NMS ----
    int t = 0;
    for (; t < TSEL; ++t) {
        // block argmax of s_sc[0..M), smallest-index tiebreak
        float bv = -3.0e38f;
        int   bj = 0x7FFFFFFF;
        for (int j = tid; j < M; j += 256) {
            float v = s_sc[j];
            if (v > bv || (v == bv && j < bj)) { bv = v; bj = j; }
        }
#pragma unroll
        for (int off = 16; off; off >>= 1) {
            float ov = __shfl_xor(bv, off, 32);
            int   oj = __shfl_xor(bj, off, 32);
            if (ov > bv || (ov == bv && oj < bj)) { bv = ov; bj = oj; }
        }
        if (lane == 0) { s_rv[wid] = bv; s_rj[wid] = bj; }
        __syncthreads();
        if (tid == 0) {
            float v = s_rv[0]; int j = s_rj[0];
#pragma unroll
            for (int w = 1; w < 8; ++w)
                if (s_rv[w] > v || (s_rv[w] == v && s_rj[w] < j)) { v = s_rv[w]; j = s_rj[w]; }
            s_selv = v; s_selj = j;
            if (v >= CONF_T) {
                s_selbox = s_box[j];
                sel_idx[outb + t]   = s_ix[j];
                sel_score[outb + t] = v;
                sel_pick[outb + t]  = 1;
            }
        }
        __syncthreads();
        if (s_selv < CONF_T) break;   // uniform: remaining rounds never pick

        float4 sb = s_selbox;
        float area_s = (sb.z - sb.x) * (sb.w - sb.y);
        int jm = s_selj;
        for (int j = tid; j < M; j += 256) {
            float4 bb = s_box[j];
            float lx = fmaxf(sb.x, bb.x), ly = fmaxf(sb.y, bb.y);
            float rx = fminf(sb.z, bb.z), ry = fminf(sb.w, bb.w);
            float iw = fmaxf(rx - lx, 0.0f), ih = fmaxf(ry - ly, 0.0f);
            float inter = iw * ih;
            float uni = area_s + (bb.z - bb.x) * (bb.w - bb.y) - inter;
            float iou = (uni > 0.0f) ? (inter / uni) : 0.0f;
            float ns;
            if (j == jm)          ns = NEGV;
            else if (iou > IOU_T) ns = NEGV;
            else                  ns = s_sc[j] * expf(scale * iou * iou);
            s_sc[j] = ns;
        }
        __syncthreads();
    }
    // rounds after early exit: pick = 0 (ignored downstream but must be defined)
    for (int tt = t + tid; tt < TSEL; tt += 256) {
        sel_pick[outb + tt]  = 0;
        sel_idx[outb + tt]   = 0;
        sel_score[outb + tt] = 0.0f;
    }
}

// ---------------------------------------------------------------------------
// Kernel 3: per-batch stable compaction / stable top-100 by score.
// Output layout (floats): [valid(4) | boxes(4*100*4) | scores(4*100) | classes(4*100)]
// ---------------------------------------------------------------------------
__global__ __launch_bounds__(256) void post_kernel(
    const float* __restrict__ boxes,      // [B, N, 4]
    const int*   __restrict__ sel_idx,
    const float* __restrict__ sel_score,
    const int*   __restrict__ sel_pick,
    float* __restrict__ out)
{
    const int M = NCLS * TSEL;            // 1600 class-major entries
    __shared__ float s_sc[NCLS * TSEL];
    __shared__ int   s_ix[NCLS * TSEL];
    __shared__ unsigned char s_va[NCLS * TSEL];
    __shared__ int s_nv;

    int b = blockIdx.x, tid = threadIdx.x;
    if (tid == 0) s_nv = 0;
    __syncthreads();

    int lc = 0;
    for (int m = tid; m < M; m += 256) {
        int g = b * M + m;
        int p = sel_pick[g];
        s_va[m] = (unsigned char)p;
        s_sc[m] = sel_score[g];
        s_ix[m] = sel_idx[g];
        lc += p;
    }
    atomicAdd(&s_nv, lc);
    __syncthreads();

    int  nv   = s_nv;
    int  vd   = (nv < MAXDET) ? nv : MAXDET;
    bool over = nv > MAXDET;

    float* out_vd = out;
    float* out_bx = out + NBATCH;
    float* out_sc = out + NBATCH + NBATCH * MAXDET * 4;
    float* out_cl = out + NBATCH + NBATCH * MAXDET * 4 + NBATCH * MAXDET;
    if (tid == 0) out_vd[b] = (float)vd;

    for (int m = tid; m < M; m += 256) {
        if (!s_va[m]) continue;
        float se = s_sc[m];
        int rank = 0;
        if (over) {
            // stable descending-score rank: higher score first, ties by position
            for (int f = 0; f < M; ++f)
                if (s_va[f] && (s_sc[f] > se || (s_sc[f] == se && f < m))) ++rank;
        } else {
            for (int f = 0; f < m; ++f) rank += s_va[f];  // compaction position
        }
        if (rank < MAXDET) {
            const float* bx = boxes + ((size_t)b * NANCH + (size_t)s_ix[m]) * 4;
            float* ob = out_bx + ((size_t)b * MAXDET + rank) * 4;
            ob[0] = bx[0]; ob[1] = bx[1]; ob[2] = bx[2]; ob[3] = bx[3];
            out_sc[b * MAXDET + rank] = se;
            out_cl[b * MAXDET + rank] = (float)(m / TSEL);
        }
    }
    // slots with no writer
    for (int s = vd + tid; s < MAXDET; s += 256) {
        float* ob = out_bx + ((size_t)b * MAXDET + s) * 4;
        ob[0] = ob[1] = ob[2] = ob[3] = 0.0f;
        out_sc[b * MAXDET + s] = 0.0f;
        out_cl[b * MAXDET + s] = -1.0f;
    }
}

// ---------------------------------------------------------------------------
extern "C" void kernel_launch(void* const* d_in, const int* in_sizes, int n_in,
                              void* d_out, int out_size, void* d_ws, size_t ws_size,
                              hipStream_t stream) {
    (void)in_sizes; (void)n_in; (void)out_size; (void)ws_size;
    const float* pred    = (const float*)d_in[0];  // [4, 49104, 20]
    const float* anchors = (const float*)d_in[1];  // [49104, 4]
    float* out = (float*)d_out;

    char* ws = (char*)d_ws;
    size_t off = 0;
    float* boxes   = (float*)(ws + off); off += (size_t)NBATCH * NANCH * 4 * sizeof(float);
    float* mscore  = (float*)(ws + off); off += (size_t)NBATCH * NANCH * sizeof(float);
    int*   mcls    = (int*)  (ws + off); off += (size_t)NBATCH * NANCH * sizeof(int);
    int*   selix   = (int*)  (ws + off); off += (size_t)NBATCH * NCLS * TSEL * sizeof(int);
    float* selsc   = (float*)(ws + off); off += (size_t)NBATCH * NCLS * TSEL * sizeof(float);
    int*   selpk   = (int*)  (ws + off); off += (size_t)NBATCH * NCLS * TSEL * sizeof(int);

    // allow ~98 KB dynamic LDS (WGP has 320 KB) — deterministic, capture-safe
    (void)hipFuncSetAttribute(reinterpret_cast<const void*>(nms_kernel),
                              hipFuncAttributeMaxDynamicSharedMemorySize,
                              (int)DYN_LDS);

    decode_kernel<<<(NBATCH * NANCH + 255) / 256, 256, 0, stream>>>(
        pred, anchors, boxes, mscore, mcls);
    nms_kernel<<<NBATCH * NCLS, 256, DYN_LDS, stream>>>(
        boxes, mscore, mcls, selix, selsc, selpk);
    post_kernel<<<NBATCH, 256, 0, stream>>>(boxes, selix, selsc, selpk, out);
}